// NestedDropout_60430189854823
// MI455X (gfx1250) — compile-verified
//
#include <hip/hip_runtime.h>

// NestedDropout: out[i,j] = X[i,j] if j < clip(geom[i]*8, 8, 512) else 0
// Pure prefix-copy + suffix-zero over [131072, 512] f32.
// Roofline: write 256 MB always; read only below cutoff (E[cutoff]=256 ->
// ~128 MB) => ~384 MB at 23.3 TB/s ~= 16.5 us. 100% HBM-bound; no matmul
// structure -> WMMA not applicable. CDNA5 levers: wave32-per-row mapping,
// EXEC-masked b128 accesses (32B-granular cutoff), async global<->LDS DMA
// (ASYNCcnt), NT temporal hints (streams exceed 192 MB L2).

typedef float v4f __attribute__((ext_vector_type(4)));
typedef int   v4i __attribute__((__vector_size__(16)));

#define FEAT 512
#define GROUP 8

// Async-builtin availability. Device pass: feature-test. Host pass: the
// builtins are not visible (__has_builtin==false there), but the host only
// needs a launch stub, so it takes the "available" path unconditionally.
#if defined(__HIP_DEVICE_COMPILE__)
# if defined(__has_builtin)
#  if __has_builtin(__builtin_amdgcn_global_load_async_to_lds_b128) && \
      __has_builtin(__builtin_amdgcn_global_store_async_from_lds_b128) && \
      __has_builtin(__builtin_amdgcn_s_wait_asynccnt)
#   define ASYNC_OK 1
#  else
#   define ASYNC_OK 0
#  endif
# else
#  define ASYNC_OK 0
# endif
#else
# define ASYNC_OK 1
#endif

#if ASYNC_OK && defined(__HIP_DEVICE_COMPILE__)
// arg0 of both async builtins is v4i addrspace(1)* (per compiler diagnostic);
// LDS side is addrspace(3).
#define AS1_V4I(p) ((__attribute__((address_space(1))) v4i*)(void*)(p))
#define AS3_V4I(p) ((__attribute__((address_space(3))) v4i*)(void*)(p))
#endif

__global__ __launch_bounds__(256) void nested_dropout_kernel(
    const float* __restrict__ X,
    const int* __restrict__ geom,
    float* __restrict__ out,
    int n_rows)
{
#if !defined(__HIP_DEVICE_COMPILE__)
  // Host stub body (host pass cannot see the device builtins).
  (void)X; (void)geom; (void)out; (void)n_rows;
#else
  const int lane          = threadIdx.x & 31;
  const int waveInBlock   = threadIdx.x >> 5;
  const int wavesPerBlock = blockDim.x >> 5;
  const int rowStride     = gridDim.x * wavesPerBlock;
  int row = blockIdx.x * wavesPerBlock + waveInBlock;

# if ASYNC_OK
  // 2 KB LDS slice per wave (16 KB/block; LDS is 320 KB/WGP -> no occupancy
  // impact). 16B-aligned for b128 LDS access.
  __shared__ __attribute__((aligned(16))) float lds[8 * FEAT];
  float* slice = &lds[waveInBlock * FEAT];

  for (; row < n_rows; row += rowStride) {
    int cutoff = geom[row] * GROUP;              // wave-uniform, multiple of 8
    cutoff = cutoff < GROUP ? GROUP : cutoff;
    cutoff = cutoff > FEAT  ? FEAT  : cutoff;

    const float* src = X   + (size_t)row * FEAT;
    float*       dst = out + (size_t)row * FEAT;

    // Previous iteration's async stores may still be reading this LDS slice.
    __builtin_amdgcn_s_wait_asynccnt(0);

    // Prefix: fire-and-forget DMA global->LDS. EXEC-masked: passes fully
    // above cutoff issue no HBM traffic (cutoff%8==0, chunks are 4 floats).
#pragma unroll
    for (int p = 0; p < FEAT / 128; ++p) {
      const int col = (p << 7) | (lane << 2);    // 0..508 step 4
      if (col < cutoff)
        __builtin_amdgcn_global_load_async_to_lds_b128(
            AS1_V4I(src + col), AS3_V4I(slice + col),
            /*offset=*/0, /*cpol=*/1 /* th:NT */);
    }

    // Suffix: stream zeros straight from VGPRs while the loads are in
    // flight (overlaps HBM read latency; no LDS involvement).
#pragma unroll
    for (int p = 0; p < FEAT / 128; ++p) {
      const int col = (p << 7) | (lane << 2);
      if (col >= cutoff) {
        v4f z = {0.0f, 0.0f, 0.0f, 0.0f};
        __builtin_nontemporal_store(z, (v4f*)(dst + col));
      }
    }

    __builtin_amdgcn_s_wait_asynccnt(0);         // prefix resident in LDS

    // Prefix: fire-and-forget DMA LDS->global.
#pragma unroll
    for (int p = 0; p < FEAT / 128; ++p) {
      const int col = (p << 7) | (lane << 2);
      if (col < cutoff)
        __builtin_amdgcn_global_store_async_from_lds_b128(
            AS1_V4I(dst + col), AS3_V4I(slice + col),
            /*offset=*/0, /*cpol=*/1 /* th:NT */);
    }
    // S_ENDPGM performs an implicit wait-idle, draining trailing async
    // stores before the wave retires.
  }
# else
  // Fallback: direct EXEC-masked b128 NT load / NT store path.
  for (; row < n_rows; row += rowStride) {
    int cutoff = geom[row] * GROUP;
    cutoff = cutoff < GROUP ? GROUP : cutoff;
    cutoff = cutoff > FEAT  ? FEAT  : cutoff;

    const v4f* src = (const v4f*)(X   + (size_t)row * FEAT);
    v4f*       dst = (v4f*)      (out + (size_t)row * FEAT);

#pragma unroll
    for (int p = 0; p < FEAT / 128; ++p) {
      const int col = (p << 7) | (lane << 2);
      const int c4  = col >> 2;
      v4f v = {0.0f, 0.0f, 0.0f, 0.0f};
      if (col < cutoff)
        v = __builtin_nontemporal_load(src + c4);
      __builtin_nontemporal_store(v, dst + c4);
    }
  }
# endif
#endif
}

extern "C" void kernel_launch(void* const* d_in, const int* in_sizes, int n_in,
                              void* d_out, int out_size, void* d_ws, size_t ws_size,
                              hipStream_t stream) {
  const float* X    = (const float*)d_in[0];   // [B, 512] f32
  const int*   geom = (const int*)  d_in[1];   // [B] i32
  float*       out  = (float*)d_out;           // [B, 512] f32

  const int n_rows        = in_sizes[1];       // B = 131072
  const int threads       = 256;               // 8 wave32's -> 8 rows/block
  const int wavesPerBlock = threads / 32;
  int blocks = (n_rows + wavesPerBlock - 1) / wavesPerBlock;  // 16384
  if (blocks < 1) blocks = 1;

  nested_dropout_kernel<<<blocks, threads, 0, stream>>>(X, geom, out, n_rows);
}